// ROVTimeRAF_35321811042727
// MI455X (gfx1250) — compile-verified
//
#include <hip/hip_runtime.h>
#include <hip/hip_bf16.h>
#include <math.h>

// ---------------------------------------------------------------------------
// Model constants (from the reference)
// ---------------------------------------------------------------------------
#define BATCH   128
#define SEQ     64
#define DIN     6
#define DMODEL  128
#define NLAYER  3
#define PDIM    24
#define NKEYS   500000
#define NHEAD   4
#define HDIM    32
#define TOPK    16
#define ROWS    (BATCH * SEQ)          // 8192
#define OUTD    (PDIM * DIN)           // 144

// Retrieval scan partitioning
#define NCHUNK      512                // key chunks (grid for scoring kernel)
#define CHUNK       1024               // keys per chunk (512*1024 >= 500000)
#define KTILE       16                 // keys staged per LDS tile
#define TILES_PER_CHUNK (CHUNK / KTILE)

// Use CDNA5 async global->LDS DMA (ASYNCcnt-tracked) for key streaming.
#define USE_ASYNC_LDS 1

typedef float v2f __attribute__((ext_vector_type(2)));
typedef float v8f __attribute__((ext_vector_type(8)));
typedef float f4  __attribute__((ext_vector_type(4)));

// ---------------------------------------------------------------------------
// Workspace layout (floats).  Total ~13.7M floats ~= 52.2 MB of d_ws.
// ---------------------------------------------------------------------------
#define OFF_H     0u                                   // [8192,128]
#define OFF_A     (OFF_H    + ROWS * DMODEL)           // [8192,128] LN out
#define OFF_QKV   (OFF_A    + ROWS * DMODEL)           // [8192,384]
#define OFF_O     (OFF_QKV  + ROWS * 3 * DMODEL)       // [8192,128] attn out
#define OFF_FC    (OFF_O    + ROWS * DMODEL)           // [8192,512]
#define OFF_BASE  (OFF_FC   + ROWS * 4 * DMODEL)       // [128,144]
#define OFF_T1    (OFF_BASE + BATCH * OUTD)            // [8192,128] retrieval embed
#define OFF_QEMB  (OFF_T1   + ROWS * DMODEL)           // [128,128]
#define OFF_CV    (OFF_QEMB + BATCH * DMODEL)          // cand vals [512*128*16]
#define OFF_CI    (OFF_CV   + NCHUNK * BATCH * TOPK)   // cand idx  [512*128*16]

#define FLAG_GELU 1
#define FLAG_RES  2

__device__ __forceinline__ float gelu_new(float x) {
    const float c = 0.7978845608028654f;               // sqrt(2/pi)
    float x3 = x * x * x;
    return 0.5f * x * (1.0f + tanhf(c * (x + 0.044715f * x3)));
}

// ---------------------------------------------------------------------------
// x[rows,6] @ W[6,128] + b  (+ wpe[s] broadcast)  -> out[rows,128]
// ---------------------------------------------------------------------------
__global__ void k_affine6(const float* __restrict__ x, const float* __restrict__ W,
                          const float* __restrict__ bias, const float* __restrict__ wpe,
                          float* __restrict__ out, int nrows) {
    int gid = blockIdx.x * blockDim.x + threadIdx.x;
    if (gid >= nrows * DMODEL) return;
    int row = gid >> 7, c = gid & 127;
    const float* xr = x + row * DIN;
    float acc = bias[c];
#pragma unroll
    for (int j = 0; j < DIN; ++j) acc += xr[j] * W[j * DMODEL + c];
    if (wpe) acc += wpe[(row & (SEQ - 1)) * DMODEL + c];   // row = b*S+s
    out[gid] = acc;
}

// ---------------------------------------------------------------------------
// LayerNorm over rows of 128: one 128-thread block per row.
// ---------------------------------------------------------------------------
__global__ void k_layernorm(const float* __restrict__ in, const float* __restrict__ g,
                            const float* __restrict__ bvec, float* __restrict__ out) {
    __shared__ float red[DMODEL];
    int r = blockIdx.x, t = threadIdx.x;
    float v = in[(size_t)r * DMODEL + t];
    red[t] = v; __syncthreads();
    for (int off = 64; off > 0; off >>= 1) { if (t < off) red[t] += red[t + off]; __syncthreads(); }
    float mu = red[0] * (1.0f / DMODEL); __syncthreads();
    float d = v - mu;
    red[t] = d * d; __syncthreads();
    for (int off = 64; off > 0; off >>= 1) { if (t < off) red[t] += red[t + off]; __syncthreads(); }
    float var = red[0] * (1.0f / DMODEL);
    float inv = 1.0f / sqrtf(var + 1e-5f);
    out[(size_t)r * DMODEL + t] = d * inv * g[t] + bvec[t];
}

// ---------------------------------------------------------------------------
// WMMA f32 GEMM: D[M,N] = A[M,K] @ B[K,N] + bias (+res) (gelu optional)
// One wave32 computes a 16x(16*NSUB) strip: NSUB accumulators share one A
// fragment (2 A-loads + 2*NSUB B-loads per K-step for NSUB WMMAs).  NSUB is a
// *template* parameter so the body is straight-line: the compiler can clause
// all loads of a K-step and issue the WMMAs back-to-back with partial
// s_wait_loadcnt, overlapping memory with the matrix pipe (the round-2
// runtime-nsub version serialized load->wait(0)->wmma).  EXEC is all-ones
// everywhere (no divergence), as WMMA requires.  M%16==0, K%4==0 always.
// A frag: lanes 0-15 -> M=0..15 (VGPR0=K+0, VGPR1=K+1); lanes 16-31 -> K+2/K+3.
// B frag: mirrored (lane=N, VGPR+half = K slot).  C/D: VGPR r = row r / r+8.
// ---------------------------------------------------------------------------
template <int NSUB>
__global__ void k_gemm_wmma_f32(const float* __restrict__ A, int lda,
                                const float* __restrict__ B, int ldb,
                                const float* __restrict__ bias,
                                const float* __restrict__ res,
                                float* __restrict__ D, int ldd,
                                int n_off, int K, int flags) {
    const int n0   = n_off + blockIdx.x * (16 * NSUB);
    const int m0   = blockIdx.y * 16;
    const int lane = threadIdx.x;          // 0..31 (one wave)
    const int half = lane >> 4;            // 0 or 1
    const int l16  = lane & 15;

    v8f acc[NSUB];
#pragma unroll
    for (int j = 0; j < NSUB; ++j) acc[j] = v8f{0.f,0.f,0.f,0.f,0.f,0.f,0.f,0.f};

    const float* arow = A + (size_t)(m0 + l16) * lda;
    const float* bcol = B + n0 + l16;

#pragma unroll 4
    for (int k = 0; k < K; k += 4) {
        const int ka = k + half * 2;
        v2f af = *(const v2f*)(arow + ka);                 // one b64 load
        const float* b0 = bcol + (size_t)ka * ldb;
        const float* b1 = bcol + (size_t)(ka + 1) * ldb;
        v2f bf[NSUB];
#pragma unroll
        for (int j = 0; j < NSUB; ++j) {                   // batch loads first
            bf[j].x = b0[16 * j];
            bf[j].y = b1[16 * j];
        }
#pragma unroll
        for (int j = 0; j < NSUB; ++j)                     // then WMMA group
            acc[j] = __builtin_amdgcn_wmma_f32_16x16x4_f32(
                false, af, false, bf[j], (short)0, acc[j], false, false);
    }

#pragma unroll
    for (int j = 0; j < NSUB; ++j) {
        const int col = n0 + 16 * j + l16;
        const float bb = bias ? bias[col] : 0.0f;
#pragma unroll
        for (int r = 0; r < 8; ++r) {
            int row = m0 + r + half * 8;
            float v = acc[j][r] + bb;
            if (flags & FLAG_GELU) v = gelu_new(v);
            if (flags & FLAG_RES)  v += res[(size_t)row * ldd + col];
            D[(size_t)row * ldd + col] = v;
        }
    }
}

// ---------------------------------------------------------------------------
// Causal attention, one block per (batch,head); thread = query position.
// Online softmax; K/V staged in LDS.
// ---------------------------------------------------------------------------
__global__ void k_attn(const float* __restrict__ qkv, float* __restrict__ o) {
    __shared__ float ks[SEQ * HDIM];
    __shared__ float vs[SEQ * HDIM];
    int bh = blockIdx.x;
    int b = bh >> 2, h = bh & 3;
    int s = threadIdx.x;                                    // 0..63
    const float* rowq = qkv + (size_t)(b * SEQ + s) * (3 * DMODEL);
    float q[HDIM];
#pragma unroll
    for (int d = 0; d < HDIM; ++d) {
        q[d]             = rowq[h * HDIM + d];
        ks[s * HDIM + d] = rowq[DMODEL     + h * HDIM + d];
        vs[s * HDIM + d] = rowq[2 * DMODEL + h * HDIM + d];
    }
    __syncthreads();

    float m = -INFINITY, l = 0.0f, acc[HDIM];
#pragma unroll
    for (int d = 0; d < HDIM; ++d) acc[d] = 0.0f;
    const float scale = 0.17677669529663687f;               // 1/sqrt(32)
    for (int j = 0; j <= s; ++j) {
        float sc = 0.0f;
#pragma unroll
        for (int d = 0; d < HDIM; ++d) sc += q[d] * ks[j * HDIM + d];
        sc *= scale;
        float nm   = fmaxf(m, sc);
        float corr = expf(m - nm);                          // exp(-inf)=0 first iter
        float p    = expf(sc - nm);
        l = l * corr + p;
#pragma unroll
        for (int d = 0; d < HDIM; ++d) acc[d] = acc[d] * corr + p * vs[j * HDIM + d];
        m = nm;
    }
    float invl = 1.0f / l;
    float* orow = o + (size_t)(b * SEQ + s) * DMODEL + h * HDIM;
#pragma unroll
    for (int d = 0; d < HDIM; ++d) orow[d] = acc[d] * invl;
}

// ---------------------------------------------------------------------------
// Row-wise L2 normalize q_emb[128,128] in place; one block per row.
// ---------------------------------------------------------------------------
__global__ void k_l2norm(float* __restrict__ q) {
    __shared__ float red[DMODEL];
    int r = blockIdx.x, t = threadIdx.x;
    float v = q[r * DMODEL + t];
    red[t] = v * v; __syncthreads();
    for (int off = 64; off > 0; off >>= 1) { if (t < off) red[t] += red[t + off]; __syncthreads(); }
    float n = sqrtf(red[0]);
    q[r * DMODEL + t] = v / fmaxf(n, 1e-12f);
}

// ---------------------------------------------------------------------------
// Fused key-scan + per-chunk top-16.
// Block = 128 threads (thread t owns batch row t). q_emb held transposed in
// LDS (conflict-free). Keys stream through a ping-pong 16-key LDS tile.
// On CDNA5 the tile is fetched with GLOBAL_LOAD_ASYNC_TO_LDS_B128 (ASYNCcnt):
// global->LDS DMA with no VGPR round-trip; each wave always issues exactly 4
// async ops per tile (OOB addresses are clamped, scores masked later), so
// `s_wait_asynccnt 4` after issuing the *next* tile guarantees (in-order
// completion) that the current tile has landed; a barrier then publishes all
// waves' DMA before compute.  This kernel IS the HBM roofline (~256 MB once).
// ---------------------------------------------------------------------------
__global__ void k_score_topk(const float* __restrict__ qemb,
                             const float* __restrict__ keys,
                             float* __restrict__ cand_v, int* __restrict__ cand_i) {
    __shared__ float q_s[DMODEL * BATCH];          // [d][b] transposed, 64 KB
    __shared__ f4    key_s[2][KTILE * 32];         // 2 x 8 KB ping-pong

    const int t     = threadIdx.x;                 // 0..127 = batch row
    const int chunk = blockIdx.x;
    const long base = (long)chunk * CHUNK;
    const f4* kp = (const f4*)keys;                // key row = 32 f4

    for (int i = t; i < BATCH * DMODEL; i += 128) {
        int r = i >> 7, c = i & 127;
        q_s[c * BATCH + r] = qemb[i];              // transpose on the way in
    }

    float tv[TOPK]; int ti[TOPK];
#pragma unroll
    for (int j = 0; j < TOPK; ++j) { tv[j] = -INFINITY; ti[j] = 0; }
    float curmin = -INFINITY;

#if USE_ASYNC_LDS
    auto issue_tile = [&](int buf, int tile) {
        long n0 = base + (long)tile * KTILE;
#pragma unroll
        for (int v = 0; v < 4; ++v) {
            int lin = v * 128 + t;                             // 0..511 f4 slots
            long n  = n0 + (lin >> 5);
            if (n > (long)(NKEYS - 1)) n = NKEYS - 1;          // clamp; masked later
            unsigned int lds_addr =
                (unsigned int)(uintptr_t)(void*)&key_s[buf][lin];
            unsigned long long ga =
                (unsigned long long)(uintptr_t)(const void*)&kp[n * 32 + (lin & 31)];
            asm volatile("global_load_async_to_lds_b128 %0, %1, off"
                         :: "v"(lds_addr), "v"(ga) : "memory");
        }
    };
#else
    auto issue_tile = [&](int buf, int tile) {
        long n0 = base + (long)tile * KTILE;
#pragma unroll
        for (int v = 0; v < 4; ++v) {
            int lin = v * 128 + t;
            long n  = n0 + (lin >> 5);
            if (n > (long)(NKEYS - 1)) n = NKEYS - 1;
            key_s[buf][lin] = __builtin_nontemporal_load(&kp[n * 32 + (lin & 31)]);
        }
    };
#endif

    issue_tile(0, 0);
    for (int tile = 0; tile < TILES_PER_CHUNK; ++tile) {
        __syncthreads();                            // prev compute on buf^1 done
        if (tile + 1 < TILES_PER_CHUNK) {
            issue_tile((tile + 1) & 1, tile + 1);   // DMA next tile
#if USE_ASYNC_LDS
            asm volatile("s_wait_asynccnt 4" ::: "memory");   // current tile landed
#endif
        } else {
#if USE_ASYNC_LDS
            asm volatile("s_wait_asynccnt 0" ::: "memory");
#endif
        }
        __syncthreads();                            // publish all waves' DMA
        const f4* kb = key_s[tile & 1];

        float acc[KTILE];
#pragma unroll
        for (int j = 0; j < KTILE; ++j) acc[j] = 0.0f;
        for (int dv = 0; dv < 32; ++dv) {           // d in groups of 4
            int d = dv * 4;
            f4 qv;
            qv.x = q_s[(d + 0) * BATCH + t];
            qv.y = q_s[(d + 1) * BATCH + t];
            qv.z = q_s[(d + 2) * BATCH + t];
            qv.w = q_s[(d + 3) * BATCH + t];
#pragma unroll
            for (int kk = 0; kk < KTILE; ++kk) {
                f4 kf = kb[kk * 32 + dv];           // broadcast across lanes
                acc[kk] += qv.x * kf.x + qv.y * kf.y + qv.z * kf.z + qv.w * kf.w;
            }
        }

        long n0 = base + (long)tile * KTILE;
#pragma unroll
        for (int kk = 0; kk < KTILE; ++kk) {
            long n  = n0 + kk;
            float s = (n < NKEYS) ? acc[kk] : -INFINITY;
            if (s > curmin) {
                bool done = false;
#pragma unroll
                for (int j = 0; j < TOPK; ++j)
                    if (!done && tv[j] == curmin) { tv[j] = s; ti[j] = (int)n; done = true; }
                curmin = tv[0];
#pragma unroll
                for (int j = 1; j < TOPK; ++j) curmin = fminf(curmin, tv[j]);
            }
        }
    }

    long o = (long)chunk * (BATCH * TOPK) + (long)t * TOPK;
#pragma unroll
    for (int j = 0; j < TOPK; ++j) { cand_v[o + j] = tv[j]; cand_i[o + j] = ti[j]; }
}

// ---------------------------------------------------------------------------
// Final merge: per batch row, reduce 512*16 candidates to global top-16,
// softmax, gather memory_values, fuse with base_pred.  One block per row.
// ---------------------------------------------------------------------------
__global__ void k_finalize(const float* __restrict__ cand_v, const int* __restrict__ cand_i,
                           const float* __restrict__ memv, const float* __restrict__ base,
                           const float* __restrict__ fw, float* __restrict__ out) {
    __shared__ float lv[NCHUNK * TOPK];            // 8192
    __shared__ int   li[NCHUNK * TOPK];
    __shared__ float rv[256]; __shared__ int ri[256];
    __shared__ float sel_v[TOPK]; __shared__ int sel_i[TOPK];

    const int b = blockIdx.x, t = threadIdx.x;     // 256 threads
    for (int i = t; i < NCHUNK * TOPK; i += 256) {
        int c = i >> 4, j = i & 15;
        long src = (long)c * (BATCH * TOPK) + (long)b * TOPK + j;
        lv[i] = cand_v[src];
        li[i] = cand_i[src];
    }
    __syncthreads();

    for (int s = 0; s < TOPK; ++s) {
        float bv = -INFINITY; int bi = -1;
        for (int i = t; i < NCHUNK * TOPK; i += 256)
            if (lv[i] > bv) { bv = lv[i]; bi = i; }
        rv[t] = bv; ri[t] = bi;
        __syncthreads();
        for (int off = 128; off > 0; off >>= 1) {
            if (t < off && rv[t + off] > rv[t]) { rv[t] = rv[t + off]; ri[t] = ri[t + off]; }
            __syncthreads();
        }
        if (t == 0) {
            int pos = ri[0];
            sel_v[s] = rv[0];
            sel_i[s] = (pos >= 0) ? li[pos] : 0;
            if (pos >= 0) lv[pos] = -INFINITY;
        }
        __syncthreads();
    }

    float m = sel_v[0];
#pragma unroll
    for (int j = 1; j < TOPK; ++j) m = fmaxf(m, sel_v[j]);
    float w[TOPK]; float sum = 0.0f;
#pragma unroll
    for (int j = 0; j < TOPK; ++j) { w[j] = expf(sel_v[j] - m); sum += w[j]; }
    float inv   = 1.0f / sum;
    float alpha = 1.0f / (1.0f + expf(-fw[0]));

    if (t < OUTD) {
        float acc = 0.0f;
#pragma unroll
        for (int j = 0; j < TOPK; ++j) acc += w[j] * memv[(size_t)sel_i[j] * OUTD + t];
        acc *= inv;
        out[(size_t)b * OUTD + t] = alpha * base[(size_t)b * OUTD + t] + (1.0f - alpha) * acc;
    }
}

// ---------------------------------------------------------------------------
// Host-side orchestration
// ---------------------------------------------------------------------------
static inline void launch_gemm(const float* A, int lda, const float* B, int ldb,
                               const float* bias, const float* res,
                               float* D, int ldd, int M, int N, int K, int flags,
                               hipStream_t stream) {
    const int nfull = N / 64;                      // 16x64 strips (NSUB=4)
    if (nfull > 0) {
        dim3 grid(nfull, M / 16);
        k_gemm_wmma_f32<4><<<grid, 32, 0, stream>>>(A, lda, B, ldb, bias, res,
                                                    D, ldd, 0, K, flags);
    }
    const int rem = (N - nfull * 64) / 16;         // 0..3 trailing 16-col tiles
    if (rem > 0) {
        dim3 grid(1, M / 16);
        const int n_off = nfull * 64;
        if (rem == 1)
            k_gemm_wmma_f32<1><<<grid, 32, 0, stream>>>(A, lda, B, ldb, bias, res,
                                                        D, ldd, n_off, K, flags);
        else if (rem == 2)
            k_gemm_wmma_f32<2><<<grid, 32, 0, stream>>>(A, lda, B, ldb, bias, res,
                                                        D, ldd, n_off, K, flags);
        else
            k_gemm_wmma_f32<3><<<grid, 32, 0, stream>>>(A, lda, B, ldb, bias, res,
                                                        D, ldd, n_off, K, flags);
    }
}

extern "C" void kernel_launch(void* const* d_in, const int* in_sizes, int n_in,
                              void* d_out, int out_size, void* d_ws, size_t ws_size,
                              hipStream_t stream) {
    const float* x       = (const float*)d_in[0];
    const float* mkeys   = (const float*)d_in[1];
    const float* mvals   = (const float*)d_in[2];
    const float* wpe     = (const float*)d_in[3];
    const float* w_in    = (const float*)d_in[4];
    const float* b_in    = (const float*)d_in[5];
    const float* ln1_g   = (const float*)d_in[6];
    const float* ln1_b   = (const float*)d_in[7];
    const float* attn_w  = (const float*)d_in[8];
    const float* attn_b  = (const float*)d_in[9];
    const float* attn_pw = (const float*)d_in[10];
    const float* attn_pb = (const float*)d_in[11];
    const float* ln2_g   = (const float*)d_in[12];
    const float* ln2_b   = (const float*)d_in[13];
    const float* fc_w    = (const float*)d_in[14];
    const float* fc_b    = (const float*)d_in[15];
    const float* mlp_pw  = (const float*)d_in[16];
    const float* mlp_pb  = (const float*)d_in[17];
    const float* lnf_g   = (const float*)d_in[18];
    const float* lnf_b   = (const float*)d_in[19];
    const float* w_out   = (const float*)d_in[20];
    const float* b_out   = (const float*)d_in[21];
    const float* ret_w1  = (const float*)d_in[22];
    const float* ret_b1  = (const float*)d_in[23];
    const float* ret_w2  = (const float*)d_in[24];
    const float* ret_b2  = (const float*)d_in[25];
    const float* fw      = (const float*)d_in[26];
    float* out = (float*)d_out;

    float* W  = (float*)d_ws;
    float* h      = W + OFF_H;
    float* bufA   = W + OFF_A;
    float* bufQKV = W + OFF_QKV;
    float* bufO   = W + OFF_O;
    float* bufFC  = W + OFF_FC;
    float* basep  = W + OFF_BASE;
    float* t1     = W + OFF_T1;
    float* qemb   = W + OFF_QEMB;
    float* candV  = W + OFF_CV;
    int*   candI  = (int*)(W + OFF_CI);

    // 1) h = x @ w_in + b_in + wpe
    k_affine6<<<(ROWS * DMODEL + 255) / 256, 256, 0, stream>>>(x, w_in, b_in, wpe, h, ROWS);

    // 2) transformer layers
    for (int l = 0; l < NLAYER; ++l) {
        k_layernorm<<<ROWS, DMODEL, 0, stream>>>(h, ln1_g + l * DMODEL, ln1_b + l * DMODEL, bufA);
        launch_gemm(bufA, DMODEL, attn_w + (size_t)l * DMODEL * 3 * DMODEL, 3 * DMODEL,
                    attn_b + l * 3 * DMODEL, nullptr, bufQKV, 3 * DMODEL,
                    ROWS, 3 * DMODEL, DMODEL, 0, stream);
        k_attn<<<BATCH * NHEAD, SEQ, 0, stream>>>(bufQKV, bufO);
        launch_gemm(bufO, DMODEL, attn_pw + (size_t)l * DMODEL * DMODEL, DMODEL,
                    attn_pb + l * DMODEL, h, h, DMODEL,
                    ROWS, DMODEL, DMODEL, FLAG_RES, stream);
        k_layernorm<<<ROWS, DMODEL, 0, stream>>>(h, ln2_g + l * DMODEL, ln2_b + l * DMODEL, bufA);
        launch_gemm(bufA, DMODEL, fc_w + (size_t)l * DMODEL * 4 * DMODEL, 4 * DMODEL,
                    fc_b + l * 4 * DMODEL, nullptr, bufFC, 4 * DMODEL,
                    ROWS, 4 * DMODEL, DMODEL, FLAG_GELU, stream);
        launch_gemm(bufFC, 4 * DMODEL, mlp_pw + (size_t)l * 4 * DMODEL * DMODEL, DMODEL,
                    mlp_pb + l * DMODEL, h, h, DMODEL,
                    ROWS, DMODEL, 4 * DMODEL, FLAG_RES, stream);
    }

    // 3) final LN + base_pred  (bufA viewed as [128, 8192] — layout is identical)
    k_layernorm<<<ROWS, DMODEL, 0, stream>>>(h, lnf_g, lnf_b, bufA);
    launch_gemm(bufA, SEQ * DMODEL, w_out, OUTD, b_out, nullptr, basep, OUTD,
                BATCH, OUTD, SEQ * DMODEL, 0, stream);

    // 4) retrieval query: t1 = x@ret_w1+b1 ; q_emb = t1.view(128,8192)@ret_w2+b2 ; l2norm
    k_affine6<<<(ROWS * DMODEL + 255) / 256, 256, 0, stream>>>(x, ret_w1, ret_b1, nullptr, t1, ROWS);
    launch_gemm(t1, SEQ * DMODEL, ret_w2, DMODEL, ret_b2, nullptr, qemb, DMODEL,
                BATCH, DMODEL, SEQ * DMODEL, 0, stream);
    k_l2norm<<<BATCH, DMODEL, 0, stream>>>(qemb);

    // 5) fused 256 MB key scan + per-chunk top-16  (async-DMA HBM-roofline kernel)
    k_score_topk<<<NCHUNK, 128, 0, stream>>>(qemb, mkeys, candV, candI);

    // 6) merge top-16, softmax, gather memory_values, fuse with base_pred
    k_finalize<<<BATCH, 256, 0, stream>>>(candV, candI, mvals, basep, fw, out);
}